// NNet_2d_962072674385
// MI455X (gfx1250) — compile-verified
//
#include <hip/hip_runtime.h>
#include <hip/hip_bf16.h>

// CDNA5 / gfx1250: grouped masked mean via one-hot GEMM on V_WMMA_F32_16X16X4_F32.
// Sums[16 segs x 16 ch] = OneHot^T (16 x K) * Rep (K x 16), K = all pixels.
// Counts via a second WMMA against an all-ones B.

typedef __attribute__((ext_vector_type(2))) float v2f;
typedef __attribute__((ext_vector_type(8))) float v8f;

constexpr int kBS   = 24;
constexpr int kC    = 16;
constexpr int kH    = 320;
constexpr int kW    = 320;
constexpr int kHW   = kH * kW;                         // 102400 (divisible by 32)
constexpr int kNPix = kBS * kHW;                       // 2,457,600

constexpr int kBlocksA    = 480;
constexpr int kWavesPerWG = 8;
constexpr int kThreads    = kWavesPerWG * 32;          // 256
constexpr int kPixPerWave = kNPix / (kBlocksA * kWavesPerWG); // 640
constexpr int kIters      = kPixPerWave / 32;          // 20
constexpr int kPartStride = 272;                       // 256 sums + 16 counts per block

__global__ __launch_bounds__(kThreads)
void proto_reduce_kernel(const float* __restrict__ rep,
                         const int* __restrict__ pmi,        // [N,4]
                         const int* __restrict__ target,     // [N]
                         const int* __restrict__ smask,      // [H*W]
                         const unsigned char* __restrict__ cond, // [N] (jnp.bool_: 1 byte/elem)
                         float* __restrict__ partial) {
  const int tid  = threadIdx.x;
  const int lane = tid & 31;
  const int wv   = tid >> 5;
  const int s    = lane & 15;   // segment row for A, channel column for B
  const int hi   = lane >> 4;   // half-wave: pixel pair (0,1) vs (2,3) of each K=4 group

  const int wave  = blockIdx.x * kWavesPerWG + wv;
  const int start = wave * kPixPerWave;

  v8f acc = {};                 // Sums[16x16] accumulator (lane l holds rows v+8*hi, col s)
  v8f cnt = {};                 // Counts broadcast across all 16 columns
  v2f bOnes; bOnes.x = 1.0f; bOnes.y = 1.0f;

  for (int it = 0; it < kIters; ++it) {
    const int base = start + it * 32;      // 32-aligned, never straddles a batch
    const int bb   = base / kHW;
    const int hw0  = base - bb * kHW;
    const int p    = base + lane;

    // Per-pixel metadata: lane = pixel within the 32-block (loaded once per 32 pixels).
    const int tgt = target[p];
    const int cv  = cond[p];
    const int sm  = smask[hw0 + lane];
    const int pm  = pmi[(size_t)p * 4 + tgt];
    const bool valid = (sm == 1) & (tgt != 0) & (cv != 0);
    const int code = valid ? (tgt * 4 + pm) : -1;   // -1 never matches s in [0,16)

    // Channel-major rep row for this lane's channel (s), batch bb.
    const float* repRow = rep + (size_t)bb * (kC * kHW) + (size_t)s * kHW + hw0;

    #pragma unroll
    for (int j = 0; j < 8; ++j) {          // 8 K=4 pixel groups per 32-block
      const int srcA = 4 * j + 2 * hi;     // first pixel of this lane's pair
      const int cA = __builtin_amdgcn_ds_bpermute(srcA << 2, code);
      const int cB = __builtin_amdgcn_ds_bpermute((srcA + 1) << 2, code);
      v2f a;
      a.x = (cA == s) ? 1.0f : 0.0f;       // A[M=s][K=2*hi+0]
      a.y = (cB == s) ? 1.0f : 0.0f;       // A[M=s][K=2*hi+1]

      const float2 bv = *(const float2*)(repRow + srcA);  // global_load_b64, streaming
      v2f b; b.x = bv.x; b.y = bv.y;       // B[K=2*hi+{0,1}][N=s]

      acc = __builtin_amdgcn_wmma_f32_16x16x4_f32(
          false, a, false, b, (short)0, acc, false, false);
      cnt = __builtin_amdgcn_wmma_f32_16x16x4_f32(
          false, a, false, bOnes, (short)0, cnt, false, false);
    }
  }

  // Deterministic cross-wave reduction in LDS (fixed summation order, no float atomics).
  __shared__ float lsum[kWavesPerWG][256];
  __shared__ float lcnt[kWavesPerWG][16];
  #pragma unroll
  for (int v = 0; v < 8; ++v)
    lsum[wv][(v + 8 * hi) * 16 + s] = acc[v];   // D[v] = Sums[v+8*hi][s]
  if (s == 0) {
    #pragma unroll
    for (int v = 0; v < 8; ++v)
      lcnt[wv][v + 8 * hi] = cnt[v];            // counts broadcast: take column 0
  }
  __syncthreads();

  float x = 0.f;
  #pragma unroll
  for (int w = 0; w < kWavesPerWG; ++w) x += lsum[w][tid];
  partial[(size_t)blockIdx.x * kPartStride + tid] = x;
  if (tid < 16) {
    float c = 0.f;
    #pragma unroll
    for (int w = 0; w < kWavesPerWG; ++w) c += lcnt[w][tid];
    partial[(size_t)blockIdx.x * kPartStride + 256 + tid] = c;
  }
}

__global__ __launch_bounds__(256)
void proto_final_kernel(const float* __restrict__ partial,
                        const float* __restrict__ protos,   // [16*16] flat
                        float* __restrict__ out) {          // [16*16] flat
  __shared__ float scnt[16];
  const int tid = threadIdx.x;

  float sum = 0.f;
  for (int g = 0; g < kBlocksA; ++g) sum += partial[g * kPartStride + tid];
  if (tid < 16) {
    float c = 0.f;
    for (int g = 0; g < kBlocksA; ++g) c += partial[g * kPartStride + 256 + tid];
    scnt[tid] = c;
  }
  __syncthreads();

  const float c    = scnt[tid >> 4];
  const float mean = sum / fmaxf(c, 1.0f);
  const float pv   = protos[tid];
  out[tid] = (c > 0.f) ? (0.99f * pv + 0.01f * mean) : pv;
}

extern "C" void kernel_launch(void* const* d_in, const int* in_sizes, int n_in,
                              void* d_out, int out_size, void* d_ws, size_t ws_size,
                              hipStream_t stream) {
  (void)in_sizes; (void)n_in; (void)out_size; (void)ws_size;
  const float*         rep    = (const float*)d_in[0];
  const int*           pmi    = (const int*)d_in[1];
  const int*           target = (const int*)d_in[2];
  const int*           smask  = (const int*)d_in[3];
  const unsigned char* cond   = (const unsigned char*)d_in[4];
  const float*         protos = (const float*)d_in[5];
  float* out     = (float*)d_out;
  float* partial = (float*)d_ws;   // 480 * 272 * 4 B = 522,240 B, fully overwritten each call

  proto_reduce_kernel<<<kBlocksA, kThreads, 0, stream>>>(rep, pmi, target, smask, cond, partial);
  proto_final_kernel<<<1, 256, 0, stream>>>(partial, protos, out);
}